// SegmentPhysicsModel_64278480552277
// MI455X (gfx1250) — compile-verified
//
#include <hip/hip_runtime.h>
#include <hip/hip_bf16.h>
#include <math.h>

// ---------------------------------------------------------------------------
// SegmentPhysicsModel: solar-panel single-diode Newton solve, elementwise.
//   N = B*L = 131072 samples, 11 float32 features each, 1 float32 output.
// Roofline: ~6.3 MB traffic (0.27us @ 23.3 TB/s) vs ~150 v_exp_f32 +
// ~150 v_rcp_f32 + ~1800 FMAs per sample -> transcendental/VALU bound.
// Strategy: carry all 6 independent diode solves through ONE Newton loop so
// each iteration issues 6 independent v_exp_f32 chains (TRANS/VALU co-exec),
// instead of 6 sequential latency-bound solves. No matmul -> WMMA N/A.
// ---------------------------------------------------------------------------

#define TB 256          // 8 wave32 waves per block
#define NFEAT 11
#define NSOLVE 6

// ---- physics constants (match reference, float32 math) --------------------
#define SOLAR_CONSTANT 1366.0f
#define SUN_EARTH_DIST 149600000.0f
#define G_REF          1366.0f
#define T_REF          298.15f
#define KB_Q           8.617333262e-05f
#define N_IDEAL        2.5f
#define JSC            170.0f
#define ALPHA_ISC      0.0006f
#define VOC_CELL       2.35f
#define RS_CELL        0.01f
#define RSH_CELL       500.0f
#define NEWTON_ITERS   25
#define INV_SERIES     (1.0f / 18.0f)     // every config uses series = 18
#define AREA_BIG       (0.0604f * 0.0398f)
#define AREA_SMALL     (0.0403f * 0.0306f)
#define VT_REF         (N_IDEAL * KB_Q * T_REF)
#define INV_RSH        (1.0f / RSH_CELL)
#define RS_OVER_RSH    (RS_CELL / RSH_CELL)

__global__ void __launch_bounds__(TB)
SegmentPhysicsModel_64278480552277_kernel(const float* __restrict__ x,
                                          const float* __restrict__ xnf,
                                          const float* __restrict__ ynf,
                                          const float* __restrict__ xpf,
                                          const float* __restrict__ ypf,
                                          float* __restrict__ out, int N) {
    __shared__ float tile[TB * NFEAT];                  // 11264 B of 320 KB LDS

    const int blockBase = blockIdx.x * TB;
    const int tid       = threadIdx.x;
    const float* src    = x + (size_t)blockBase * NFEAT;

    // Speculative gfx1250 prefetch of the NEXT block's tile
    // (global_prefetch_b8); OOB speculative prefetch is silently dropped.
    __builtin_prefetch(src + (size_t)TB * NFEAT, 0, 0);

    // Coalesced global -> LDS staging.
    if (blockBase + TB <= N) {
        // Full block: compile-time trip count -> straight-line 11 loads.
        #pragma unroll
        for (int k = 0; k < NFEAT; ++k)
            tile[tid + k * TB] = src[tid + k * TB];
    } else {
        const int nelts = (N - blockBase) * NFEAT;
        for (int j = tid; j < nelts; j += TB)
            tile[j] = src[j];
    }
    __syncthreads();

    if (blockBase + tid >= N) return;

    // Stride-11 LDS reads: 11 coprime with 64 banks -> conflict-free in wave32.
    const float* r = &tile[tid * NFEAT];
    const float sun_x      = r[0];
    const float sun_y      = r[1];
    const float temp_pos_x = r[2];
    const float temp_neg_x = r[3];
    const float temp_pos_y = r[4];
    const float temp_neg_y = r[5];
    const float voltage    = r[6];
    const float distance   = r[7];
    // r[8..10] = year/month/day: unused by the reference math.

    const float ratio      = SUN_EARTH_DIST / distance;
    const float irradiance = SOLAR_CONSTANT * ratio * ratio;
    const float v_cell     = voltage * INV_SERIES;

    // Uniform constant; clang constant-folds expm1f(literal) at -O3.
    const float inv_em1 = 1.0f / expm1f(VOC_CELL / VT_REF);

    const float g_xp = irradiance * fmaxf( sun_x, 0.0f);
    const float g_xn = irradiance * fmaxf(-sun_x, 0.0f);
    const float g_yp = irradiance * fmaxf( sun_y, 0.0f);
    const float g_yn = irradiance * fmaxf(-sun_y, 0.0f);

    // Six sub-config solves, interleaved for TRANS-pipe ILP:
    //  0: Y_NEG big   (108/18=6)   1: X_NEG big (36/18=2)  2: X_NEG small (126/18=7)
    //  3: X_POS big   (2)          4: X_POS small (7)      5: Y_POS small (180/18=10)
    const float g_s[NSOLVE]    = { g_yn, g_xn, g_xn, g_xp, g_xp, g_yp };
    const float T_s[NSOLVE]    = { temp_neg_y, temp_neg_x, temp_neg_x,
                                   temp_pos_x, temp_pos_x, temp_pos_y };
    const float area_s[NSOLVE] = { AREA_BIG, AREA_BIG, AREA_SMALL,
                                   AREA_BIG, AREA_SMALL, AREA_SMALL };

    float iph[NSOLVE], inv_vt[NSOLVE], i0c[NSOLVE], ae[NSOLVE], I[NSOLVE];
    #pragma unroll
    for (int s = 0; s < NSOLVE; ++s) {
        iph[s]    = JSC * area_s[s] * (g_s[s] * (1.0f / G_REF)) *
                    fmaf(ALPHA_ISC, T_s[s] - T_REF, 1.0f);
        const float vt = N_IDEAL * KB_Q * T_s[s];
        inv_vt[s] = 1.0f / vt;                    // exact; hoisted out of loop
        i0c[s]    = JSC * area_s[s] * inv_em1;
        ae[s]     = i0c[s] * RS_CELL * inv_vt[s]; // e-coefficient of -f'
        I[s]      = iph[s];
    }

    // One Newton loop, 6 independent exp/rcp chains per iteration.
    #pragma unroll 1
    for (int k = 0; k < NEWTON_ITERS; ++k) {
        #pragma unroll
        for (int s = 0; s < NSOLVE; ++s) {
            const float t = fmaf(I[s], RS_CELL, v_cell);        // v_cell + I*Rs
            float z = t * inv_vt[s];
            z = fminf(fmaxf(z, -40.0f), 40.0f);                 // reference clip
            const float e = __expf(z);                          // v_exp_f32
            // f  = iph - i0*(e-1) - t/Rsh - I
            const float f = iph[s] + i0c[s]
                          - fmaf(i0c[s], e, fmaf(t, INV_RSH, I[s]));
            // fp = -(ae*e + Rs/Rsh + 1);  I -= f/fp  ==  I += f/negfp
            const float negfp = fmaf(ae[s], e, 1.0f + RS_OVER_RSH);
            I[s] = fmaf(f, __frcp_rn(negfp), I[s]);             // v_rcp_f32
        }
    }

    // Weighted sum: panel factors then global 0.92 derate.
    const float wxn = xnf[0], wyn = ynf[0], wxp = xpf[0], wyp = ypf[0];
    const float par_s[NSOLVE] = { 6.0f, 2.0f, 7.0f, 2.0f, 7.0f, 10.0f };
    const float w_s[NSOLVE]   = { wyn, wxn, wxn, wxp, wxp, wyp };

    float total = 0.0f;
    #pragma unroll
    for (int s = 0; s < NSOLVE; ++s)
        total = fmaf(par_s[s] * fmaxf(I[s], 0.0f), w_s[s], total);

    out[blockBase + tid] = total * 0.92f;
}

extern "C" void kernel_launch(void* const* d_in, const int* in_sizes, int n_in,
                              void* d_out, int out_size, void* d_ws, size_t ws_size,
                              hipStream_t stream) {
    const float* x   = (const float*)d_in[0];
    const float* xnf = (const float*)d_in[1];
    const float* ynf = (const float*)d_in[2];
    const float* xpf = (const float*)d_in[3];
    const float* ypf = (const float*)d_in[4];
    float* out = (float*)d_out;

    const int N = in_sizes[0] / NFEAT;   // B*L samples
    const int grid = (N + TB - 1) / TB;

    SegmentPhysicsModel_64278480552277_kernel<<<grid, TB, 0, stream>>>(
        x, xnf, ynf, xpf, ypf, out, N);
}